// ToBEVConvolution_42279658062069
// MI455X (gfx1250) — compile-verified
//
#include <hip/hip_runtime.h>

typedef __attribute__((ext_vector_type(2))) float v2f;
typedef __attribute__((ext_vector_type(8))) float v8f;

namespace {
constexpr int CIN  = 32;
constexpr int COUT = 64;
constexpr int KNUM = 16;
constexpr int GDIM = 512;
constexpr int BDIM = 2;

// d_ws layout (in int32 units)
constexpr int WS_COUNTS = 0;   // 16: per-h point counts
constexpr int WS_OFF    = 16;  // 17: exclusive point offsets per bucket
constexpr int WS_TOFF   = 33;  // 17: exclusive 16-point tile offsets per bucket
constexpr int WS_CUR    = 50;  // 16: scatter cursors
constexpr int WS_IDX    = 66;  // N : point indices grouped by bucket
}

// ---- Pass 1: zero the 128MB output grid (single write pass) + counters ----
__global__ void zero_grid_kernel(float4* __restrict__ out, int n4,
                                 int* __restrict__ counts) {
    int i = blockIdx.x * blockDim.x + threadIdx.x;
    if (i < n4) out[i] = make_float4(0.f, 0.f, 0.f, 0.f);
    if (blockIdx.x == 0 && threadIdx.x < KNUM) counts[threadIdx.x] = 0;
}

// ---- Pass 2: histogram points by kernel slice ----
__global__ void count_kernel(const int* __restrict__ coords,
                             const int* __restrict__ tsp, int n,
                             int* __restrict__ counts) {
    int i = blockIdx.x * blockDim.x + threadIdx.x;
    if (i >= n) return;
    int h = coords[i * 4 + 1] / tsp[0];
    h = min(max(h, 0), KNUM - 1);
    atomicAdd(&counts[h], 1);
}

// ---- Pass 3: serial scan over 16 buckets (trivial) ----
__global__ void scan_kernel(int* __restrict__ ws) {
    if (threadIdx.x == 0 && blockIdx.x == 0) {
        int o = 0, t = 0;
        for (int k = 0; k < KNUM; ++k) {
            ws[WS_OFF + k]  = o;
            ws[WS_TOFF + k] = t;
            ws[WS_CUR + k]  = o;
            o += ws[WS_COUNTS + k];
            t += (ws[WS_COUNTS + k] + 15) / 16;
        }
        ws[WS_OFF + KNUM]  = o;
        ws[WS_TOFF + KNUM] = t;
    }
}

// ---- Pass 4: compact point indices per bucket ----
__global__ void scatter_idx_kernel(const int* __restrict__ coords,
                                   const int* __restrict__ tsp, int n,
                                   int* __restrict__ ws) {
    int i = blockIdx.x * blockDim.x + threadIdx.x;
    if (i >= n) return;
    int h = coords[i * 4 + 1] / tsp[0];
    h = min(max(h, 0), KNUM - 1);
    int p = atomicAdd(&ws[WS_CUR + h], 1);
    ws[WS_IDX + p] = i;
}

// ---- Pass 5: one wave per 16-point tile; WMMA f32 GEMM + atomic scatter ----
__global__ void __launch_bounds__(128)
gemm_scatter_kernel(const float* __restrict__ feats, const float* __restrict__ wk,
                    const int* __restrict__ coords, const int* __restrict__ ws,
                    float* __restrict__ out) {
    const int* off  = ws + WS_OFF;
    const int* toff = ws + WS_TOFF;
    const int* idx  = ws + WS_IDX;

    int wave = blockIdx.x * (blockDim.x >> 5) + (threadIdx.x >> 5);
    int lane = threadIdx.x & 31;
    if (wave >= toff[KNUM]) return;   // wave-uniform exit (EXEC stays all-1s)

    // find bucket: last k with toff[k] <= wave
    int kb = 0;
    #pragma unroll
    for (int k = 1; k < KNUM; ++k)
        if (wave >= toff[k]) kb = k;

    int s   = off[kb] + (wave - toff[kb]) * 16;   // first point of this tile
    int end = off[kb + 1];

    const bool hi  = lane >= 16;
    const int  m16 = lane & 15;

    // A-matrix row handled by this lane (ISA 7.12.2: lanes 0-15 M=0..15, K=0/1;
    // lanes 16-31 same M, K=2/3). Ragged tail rows use row 0 as a safe address
    // and are masked to zero in registers (no EXEC-predicated loads).
    bool rv = (s + m16) < end;
    int  r  = rv ? idx[s + m16] : 0;

    // This lane's 16 A elements are 8 aligned float2 pairs, stride 16B.
    const v2f* __restrict__ fr2 =
        (const v2f*)feats + (long)r * (CIN / 2) + (hi ? 1 : 0);
    v2f a[8];
    #pragma unroll
    for (int kq = 0; kq < 8; ++kq) {
        v2f t = fr2[kq * 2];          // unconditional global_load_b64, clause'd
        a[kq].x = rv ? t.x : 0.f;     // v_cndmask, no branches
        a[kq].y = rv ? t.y : 0.f;
    }

    const float* __restrict__ w = wk + (long)kb * (CIN * COUT);

    v8f acc[4] = {};   // 4 column tiles of 16 -> COUT=64

    #pragma unroll
    for (int kq = 0; kq < 8; ++kq) {          // K=32 in steps of 4
        int kk = kq * 4 + (hi ? 2 : 0);
        const float* w0 = w + kk * COUT;      // B rows kk, kk+1 (this lane half)
        #pragma unroll
        for (int nt = 0; nt < 4; ++nt) {
            int n = nt * 16 + m16;            // B: N = lane (mod 16)
            v2f B;
            B.x = w0[n];
            B.y = w0[COUT + n];
            acc[nt] = __builtin_amdgcn_wmma_f32_16x16x4_f32(
                false, a[kq], false, B, (short)0, acc[nt], false, false);
        }
    }

    // grid base address for this lane's row (lanes 0..15 authoritative)
    int x = coords[r * 4 + 0];
    int z = coords[r * 4 + 2];
    int b = coords[r * 4 + 3];
    int baseAddr = ((x * GDIM + z) * BDIM + b) * COUT;

    // D layout: lane holds D[m][n], m = vgpr + (hi ? 8 : 0), n = nt*16 + m16.
    #pragma unroll
    for (int j = 0; j < 8; ++j) {
        int m  = j + (hi ? 8 : 0);
        int ob = __shfl(baseAddr, m, 32);     // ds_bpermute broadcast of row base
        if (s + m < end) {
            #pragma unroll
            for (int nt = 0; nt < 4; ++nt)
                atomicAdd(&out[ob + nt * 16 + m16], acc[nt][j]);
        }
    }
}

extern "C" void kernel_launch(void* const* d_in, const int* in_sizes, int n_in,
                              void* d_out, int out_size, void* d_ws, size_t ws_size,
                              hipStream_t stream) {
    const float* feats = (const float*)d_in[0];
    const float* wk    = (const float*)d_in[1];
    const int*   coords= (const int*)d_in[2];
    const int*   tsp   = (const int*)d_in[3];
    float* out = (float*)d_out;
    int*   ws  = (int*)d_ws;

    int n  = in_sizes[0] / CIN;       // number of points
    int n4 = out_size / 4;            // grid size in float4

    zero_grid_kernel<<<(n4 + 255) / 256, 256, 0, stream>>>((float4*)out, n4, ws + WS_COUNTS);
    count_kernel<<<(n + 255) / 256, 256, 0, stream>>>(coords, tsp, n, ws + WS_COUNTS);
    scan_kernel<<<1, 32, 0, stream>>>(ws);
    scatter_idx_kernel<<<(n + 255) / 256, 256, 0, stream>>>(coords, tsp, n, ws);

    int maxTiles = (n + 15) / 16 + KNUM;      // upper bound incl. per-bucket padding
    int blocks   = (maxTiles + 3) / 4;        // 4 waves (tiles) per 128-thread block
    gemm_scatter_kernel<<<blocks, 128, 0, stream>>>(feats, wk, coords, ws, out);
}